// SAGEConv_62191126446612
// MI455X (gfx1250) — compile-verified
//
#include <hip/hip_runtime.h>
#include <hip/hip_bf16.h>

#define DIM 128
#define LDA 132   // padded LDS row stride (floats) for 128-wide tiles
#define LDC 260   // padded LDS row stride (floats) for 256-wide concat tile

typedef float v2f __attribute__((ext_vector_type(2)));
typedef float v8f __attribute__((ext_vector_type(8)));

// Fast single-instruction transcendental helpers (v_rcp_f32 / v_exp_f32 / v_rsq_f32)
__device__ __forceinline__ float fast_rcp(float x) { return __builtin_amdgcn_rcpf(x); }
__device__ __forceinline__ float sigmoid_f(float v) {
    return fast_rcp(1.0f + __expf(-v));
}
__device__ __forceinline__ float silu_f(float v) {
    return v * sigmoid_f(v);
}

// D = A(16x4) x B(4x16) + C, fp32 WMMA. A,B = 2 VGPRs/lane, C/D = 8 VGPRs/lane.
__device__ __forceinline__ v8f wmma_f32_k4(v2f a, v2f b, v8f c) {
    return __builtin_amdgcn_wmma_f32_16x16x4_f32(
        /*neg_a=*/false, a, /*neg_b=*/false, b,
        /*c_mod=*/(short)0, c, /*reuse_a=*/false, /*reuse_b=*/false);
}

// ---------------------------------------------------------------------------
// Kernel 0: zero the scatter accumulator (d_out) and degree counts.
// ---------------------------------------------------------------------------
__global__ void __launch_bounds__(256) zero_kernel(float* __restrict__ out,
                                                   float* __restrict__ cnt,
                                                   int nout, int ncnt) {
    int i = blockIdx.x * blockDim.x + threadIdx.x;
    int stride = gridDim.x * blockDim.x;
    for (int j = i; j < nout; j += stride) out[j] = 0.0f;
    for (int j = i; j < ncnt; j += stride) cnt[j] = 0.0f;
}

// ---------------------------------------------------------------------------
// Kernel 1: hagg[n] = silu(x[n] @ agg_W + agg_b)   (N x 128) @ (128 x 128)
// One block = 16 rows, 8 waves; wave w owns output cols [16w, 16w+16).
// fp32 WMMA 16x16x4, A slab staged in LDS (padded), B streamed from global.
// ---------------------------------------------------------------------------
__global__ void __launch_bounds__(256) agg_gemm_kernel(
    const float* __restrict__ x, const float* __restrict__ W,
    const float* __restrict__ bias, float* __restrict__ hagg) {
    __shared__ float As[16 * LDA];
    const int rowbase = blockIdx.x * 16;
    const int t = threadIdx.x;

    // Stage 16x128 A slab (coalesced float4 loads)
    {
        int r = t >> 4, c0 = (t & 15) * 8;
        const float4* src = (const float4*)(x + (size_t)(rowbase + r) * DIM + c0);
        float4 v0 = src[0], v1 = src[1];
        float* dr = As + r * LDA + c0;
        dr[0] = v0.x; dr[1] = v0.y; dr[2] = v0.z; dr[3] = v0.w;
        dr[4] = v1.x; dr[5] = v1.y; dr[6] = v1.z; dr[7] = v1.w;
    }
    __syncthreads();

    const int wave = t >> 5, lane = t & 31;
    const int n  = (wave << 4) + (lane & 15);   // output column
    const int g2 = (lane >> 4) << 1;            // K sub-offset: 0 or 2
    const float* arow = As + (lane & 15) * LDA;

    v8f c;
    float bv = bias[n];
#pragma unroll
    for (int i = 0; i < 8; i++) c[i] = bv;

#pragma unroll 4
    for (int k0 = 0; k0 < DIM; k0 += 4) {
        int k = k0 + g2;
        v2f a = *(const v2f*)(arow + k);        // ds_load_b64, conflict-free
        v2f b;
        b.x = W[(size_t)k * DIM + n];
        b.y = W[(size_t)(k + 1) * DIM + n];
        c = wmma_f32_k4(a, b, c);
    }

    // silu + store (C layout: VGPR i -> M = i + 8*(lane>>4), N = n)
    const int m0 = rowbase + ((lane >> 4) << 3);
#pragma unroll
    for (int i = 0; i < 8; i++) {
        hagg[(size_t)(m0 + i) * DIM + n] = silu_f(c[i]);
    }
}

// ---------------------------------------------------------------------------
// Kernel 2a: per-edge gate  e_w = sigmoid(silu(ea@W1+b1)@W2+b2); degree count.
// One thread per edge; 3KB of MLP weights staged in LDS (broadcast reads).
// ---------------------------------------------------------------------------
__global__ void __launch_bounds__(256) edge_gate_kernel(
    const float* __restrict__ edge_attr, const int* __restrict__ edge_index,
    const float* __restrict__ W1, const float* __restrict__ b1,
    const float* __restrict__ W2, const float* __restrict__ b2,
    float* __restrict__ e_w, float* __restrict__ cnt, int E_) {
    __shared__ float sW1[4 * DIM];
    __shared__ float sb1[DIM];
    __shared__ float sW2[DIM];
    const int t = threadIdx.x;
    for (int i = t; i < 4 * DIM; i += 256) sW1[i] = W1[i];
    if (t < DIM) { sb1[t] = b1[t]; sW2[t] = W2[t]; }
    __syncthreads();

    int e = blockIdx.x * 256 + t;
    if (e >= E_) return;

    float4 ea = *(const float4*)(edge_attr + (size_t)e * 4);
    float acc = 0.0f;
#pragma unroll 8
    for (int j = 0; j < DIM; j++) {
        float h = sb1[j];
        h = fmaf(ea.x, sW1[j], h);
        h = fmaf(ea.y, sW1[DIM + j], h);
        h = fmaf(ea.z, sW1[2 * DIM + j], h);
        h = fmaf(ea.w, sW1[3 * DIM + j], h);
        acc = fmaf(silu_f(h), sW2[j], acc);
    }
    e_w[e] = sigmoid_f(acc + b2[0]);
    atomicAdd(&cnt[edge_index[E_ + e]], 1.0f);
}

// ---------------------------------------------------------------------------
// Kernel 2b: gated gather/scatter. One wave32 per edge at a time:
// lane l moves float4 [4l,4l+4) of the 128-float message. Coalesced 512B
// gather from hagg + 128 f32 L2 atomics per edge. Memory-bound phase.
// ---------------------------------------------------------------------------
__global__ void __launch_bounds__(256) scatter_kernel(
    const float* __restrict__ hagg, const int* __restrict__ edge_index,
    const float* __restrict__ e_w, float* __restrict__ out_sum, int E_) {
    const int lane = threadIdx.x & 31;
    const int wid = (blockIdx.x * blockDim.x + threadIdx.x) >> 5;
    const int nw  = (gridDim.x * blockDim.x) >> 5;
    for (int e = wid; e < E_; e += nw) {
        int s = edge_index[e];          // broadcast load
        int d = edge_index[E_ + e];
        float w = e_w[e];
        float4 v = *(const float4*)(hagg + (size_t)s * DIM + lane * 4);
        float* o = out_sum + (size_t)d * DIM + lane * 4;
        atomicAdd(o + 0, v.x * w);
        atomicAdd(o + 1, v.y * w);
        atomicAdd(o + 2, v.z * w);
        atomicAdd(o + 3, v.w * w);
    }
}

// ---------------------------------------------------------------------------
// Kernel 3: per 16-row slab, fully fused:
//   agg = out_sum/max(cnt,1); A = [x | agg] (16x256 in LDS)
//   h1 = silu(A @ upd_W1 + b1)           (K=256, fp32 WMMA)
//   h2 = h1 @ upd_W2 + b2 + x @ res_W    (K=128 + K=128, same accumulator)
//   out = LayerNorm(h2) * g + b
// `out` holds out_sum on entry, final result on exit (slab-local RAW is safe).
// ---------------------------------------------------------------------------
__global__ void __launch_bounds__(256) update_kernel(
    const float* __restrict__ x, const float* __restrict__ cnt,
    const float* __restrict__ W1, const float* __restrict__ b1,
    const float* __restrict__ W2, const float* __restrict__ b2,
    const float* __restrict__ resW,
    const float* __restrict__ ln_g, const float* __restrict__ ln_b,
    float* __restrict__ out) {
    __shared__ float As[16 * LDC];     // concat slab, later reused for h2
    __shared__ float Hs[16 * LDA];     // h1 slab
    __shared__ float red[2][16][17];   // layernorm partials
    const int rowbase = blockIdx.x * 16;
    const int t = threadIdx.x;

    // Stage [x | out_sum/max(cnt,1)]
    {
        int r = t >> 4, c0 = (t & 15) * 8;
        int row = rowbase + r;
        const float4* sx = (const float4*)(x + (size_t)row * DIM + c0);
        float4 v0 = sx[0], v1 = sx[1];
        float* dr = As + r * LDC + c0;
        dr[0] = v0.x; dr[1] = v0.y; dr[2] = v0.z; dr[3] = v0.w;
        dr[4] = v1.x; dr[5] = v1.y; dr[6] = v1.z; dr[7] = v1.w;
        float inv = fast_rcp(fmaxf(cnt[row], 1.0f));
        const float4* so = (const float4*)(out + (size_t)row * DIM + c0);
        float4 u0 = so[0], u1 = so[1];
        float* d2 = As + r * LDC + DIM + c0;
        d2[0] = u0.x * inv; d2[1] = u0.y * inv; d2[2] = u0.z * inv; d2[3] = u0.w * inv;
        d2[4] = u1.x * inv; d2[5] = u1.y * inv; d2[6] = u1.z * inv; d2[7] = u1.w * inv;
    }
    __syncthreads();

    const int wave = t >> 5, lane = t & 31;
    const int n  = (wave << 4) + (lane & 15);
    const int g2 = (lane >> 4) << 1;
    const int m0 = (lane >> 4) << 3;
    const float* arow = As + (lane & 15) * LDC;

    // GEMM1: K = 256
    v8f c;
    {
        float bv = b1[n];
#pragma unroll
        for (int i = 0; i < 8; i++) c[i] = bv;
    }
#pragma unroll 4
    for (int k0 = 0; k0 < 2 * DIM; k0 += 4) {
        int k = k0 + g2;
        v2f a = *(const v2f*)(arow + k);
        v2f b;
        b.x = W1[(size_t)k * DIM + n];
        b.y = W1[(size_t)(k + 1) * DIM + n];
        c = wmma_f32_k4(a, b, c);
    }
    // silu -> Hs
#pragma unroll
    for (int i = 0; i < 8; i++) Hs[(m0 + i) * LDA + n] = silu_f(c[i]);
    __syncthreads();

    // GEMM2: h1 @ W2  +  x @ resW, shared accumulator
    v8f c2;
    {
        float bv = b2[n];
#pragma unroll
        for (int i = 0; i < 8; i++) c2[i] = bv;
    }
    const float* hrow = Hs + (lane & 15) * LDA;
#pragma unroll 4
    for (int k0 = 0; k0 < DIM; k0 += 4) {
        int k = k0 + g2;
        v2f a = *(const v2f*)(hrow + k);
        v2f b;
        b.x = W2[(size_t)k * DIM + n];
        b.y = W2[(size_t)(k + 1) * DIM + n];
        c2 = wmma_f32_k4(a, b, c2);
    }
#pragma unroll 4
    for (int k0 = 0; k0 < DIM; k0 += 4) {
        int k = k0 + g2;
        v2f a = *(const v2f*)(arow + k);   // x part of concat slab
        v2f b;
        b.x = resW[(size_t)k * DIM + n];
        b.y = resW[(size_t)(k + 1) * DIM + n];
        c2 = wmma_f32_k4(a, b, c2);
    }
    __syncthreads();   // all waves finished reading As -> safe to reuse

    // h2 tile -> LDS for cross-wave LayerNorm
#pragma unroll
    for (int i = 0; i < 8; i++) As[(m0 + i) * LDC + n] = c2[i];
    __syncthreads();

    // LayerNorm: thread t handles (row = t/16, 8 cols)
    {
        int r = t >> 4, sub = t & 15, c0 = sub * 8;
        const float* hr = As + r * LDC + c0;
        float s = 0.0f, ss = 0.0f;
#pragma unroll
        for (int j = 0; j < 8; j++) { float v = hr[j]; s += v; ss += v * v; }
        red[0][r][sub] = s;
        red[1][r][sub] = ss;
        __syncthreads();
        float mu = 0.0f, m2 = 0.0f;
#pragma unroll
        for (int j = 0; j < 16; j++) { mu += red[0][r][j]; m2 += red[1][r][j]; }
        mu *= (1.0f / DIM);
        m2 = m2 * (1.0f / DIM) - mu * mu;
        float rs = __builtin_amdgcn_rsqf(m2 + 1e-5f);
        float4 o0, o1;
        o0.x = (hr[0] - mu) * rs * ln_g[c0 + 0] + ln_b[c0 + 0];
        o0.y = (hr[1] - mu) * rs * ln_g[c0 + 1] + ln_b[c0 + 1];
        o0.z = (hr[2] - mu) * rs * ln_g[c0 + 2] + ln_b[c0 + 2];
        o0.w = (hr[3] - mu) * rs * ln_g[c0 + 3] + ln_b[c0 + 3];
        o1.x = (hr[4] - mu) * rs * ln_g[c0 + 4] + ln_b[c0 + 4];
        o1.y = (hr[5] - mu) * rs * ln_g[c0 + 5] + ln_b[c0 + 5];
        o1.z = (hr[6] - mu) * rs * ln_g[c0 + 6] + ln_b[c0 + 6];
        o1.w = (hr[7] - mu) * rs * ln_g[c0 + 7] + ln_b[c0 + 7];
        float4* dst = (float4*)(out + (size_t)(rowbase + r) * DIM + c0);
        dst[0] = o0;
        dst[1] = o1;
    }
}

// ---------------------------------------------------------------------------
extern "C" void kernel_launch(void* const* d_in, const int* in_sizes, int n_in,
                              void* d_out, int out_size, void* d_ws, size_t ws_size,
                              hipStream_t stream) {
    const float* x          = (const float*)d_in[0];
    const int*   edge_index = (const int*)d_in[1];
    const float* edge_attr  = (const float*)d_in[2];
    const float* agg_W      = (const float*)d_in[3];
    const float* agg_b      = (const float*)d_in[4];
    const float* eW1        = (const float*)d_in[5];
    const float* eb1        = (const float*)d_in[6];
    const float* eW2        = (const float*)d_in[7];
    const float* eb2        = (const float*)d_in[8];
    const float* uW1        = (const float*)d_in[9];
    const float* ub1        = (const float*)d_in[10];
    const float* uW2        = (const float*)d_in[11];
    const float* ub2        = (const float*)d_in[12];
    const float* ln_g       = (const float*)d_in[13];
    const float* ln_b       = (const float*)d_in[14];
    const float* resW       = (const float*)d_in[15];

    const int Nn = in_sizes[0] / DIM;   // 100000 (divisible by 16)
    const int Ee = in_sizes[2] / 4;     // 1600000

    float* out  = (float*)d_out;        // doubles as scatter accumulator
    float* ws   = (float*)d_ws;
    float* hagg = ws;                               // N*128 floats
    float* e_w  = hagg + (size_t)Nn * DIM;          // E floats
    float* cnt  = e_w + Ee;                         // N floats

    zero_kernel<<<2048, 256, 0, stream>>>(out, cnt, Nn * DIM, Nn);
    agg_gemm_kernel<<<Nn / 16, 256, 0, stream>>>(x, agg_W, agg_b, hagg);
    edge_gate_kernel<<<(Ee + 255) / 256, 256, 0, stream>>>(
        edge_attr, edge_index, eW1, eb1, eW2, eb2, e_w, cnt, Ee);
    scatter_kernel<<<2048, 256, 0, stream>>>(hagg, edge_index, e_w, out, Ee);
    update_kernel<<<Nn / 16, 256, 0, stream>>>(
        x, cnt, uW1, ub1, uW2, ub2, resW, ln_g, ln_b, out);
}